// GAT_44014824849928
// MI455X (gfx1250) — compile-verified
//
#include <hip/hip_runtime.h>
#include <math.h>

typedef __attribute__((ext_vector_type(2))) float v2f;
typedef __attribute__((ext_vector_type(8))) float v8f;

#define FDIM 36
#define NHEAD 3
#define HDIM 12
#define NEG_SLOPE 0.2f
#define NEG_INF_KEY 0x007FFFFFu   // f2key(-inf)

__device__ __forceinline__ unsigned f2key(float f) {
  unsigned u = __float_as_uint(f);
  return (u & 0x80000000u) ? ~u : (u | 0x80000000u);
}
__device__ __forceinline__ float key2f(unsigned k) {
  return (k & 0x80000000u) ? __uint_as_float(k & 0x7FFFFFFFu)
                           : __uint_as_float(~k);
}
__device__ __forceinline__ float leaky(float x) {
  return x >= 0.0f ? x : NEG_SLOPE * x;
}

// out(nrows x 36) = A(nrows x 36) * W(36 x 36) [+ bias], fp32 WMMA.
// One wave per 16-row stripe; 3 column tiles (48 cols, 36 valid) kept in
// 3 accumulators -> 27 V_WMMA_F32_16X16X4_F32 per wave, A-frag reused 3x.
// Out-of-range rows/cols are CLAMPED (not zeroed): garbage rows/cols of the
// product are simply never stored, so no divergent predication is needed.
__global__ __launch_bounds__(32) void gemm36_wmma(
    const float* __restrict__ A, const float* __restrict__ W,
    const float* __restrict__ bias, float* __restrict__ out, int nrows) {
  const int lane = threadIdx.x;       // 0..31 (wave32)
  const int half = lane >> 4;         // 0 or 1
  const int idx  = lane & 15;

  int mrow = blockIdx.x * 16 + idx;
  mrow = mrow < nrows ? mrow : (nrows - 1);          // clamp into valid memory
  // A-frag base: lane reads A[mrow][k0+2*half + {0,1}] -> b64 at imm offset
  const float* __restrict__ Ab = A + (size_t)mrow * FDIM + 2 * half;

  // B-frag bases per column tile: element = Wb[k0*36] / Wb[(k0+1)*36]
  const int col0 = idx;
  const int col1 = idx + 16;
  const int col2 = (idx + 32) < FDIM ? (idx + 32) : (FDIM - 1);  // clamp
  const float* __restrict__ Wb0 = W + col0 + 72 * half;  // 72 = 2*FDIM
  const float* __restrict__ Wb1 = W + col1 + 72 * half;
  const float* __restrict__ Wb2 = W + col2 + 72 * half;

  v8f acc[3];
  acc[0] = (v8f){0.f,0.f,0.f,0.f,0.f,0.f,0.f,0.f};
  acc[1] = acc[0];
  acc[2] = acc[0];

  #pragma unroll
  for (int k0 = 0; k0 < FDIM; k0 += 4) {
    v2f a, b0, b1, b2;
    a.x  = Ab[k0];                 // contiguous pair -> global_load_b64
    a.y  = Ab[k0 + 1];
    b0.x = Wb0[(k0 + 0) * FDIM];   // immediate-offset b32 loads, no addr VALU
    b0.y = Wb0[(k0 + 1) * FDIM];
    b1.x = Wb1[(k0 + 0) * FDIM];
    b1.y = Wb1[(k0 + 1) * FDIM];
    b2.x = Wb2[(k0 + 0) * FDIM];
    b2.y = Wb2[(k0 + 1) * FDIM];
    acc[0] = __builtin_amdgcn_wmma_f32_16x16x4_f32(false, a, false, b0,
                                                   (short)0, acc[0], false, false);
    acc[1] = __builtin_amdgcn_wmma_f32_16x16x4_f32(false, a, false, b1,
                                                   (short)0, acc[1], false, false);
    acc[2] = __builtin_amdgcn_wmma_f32_16x16x4_f32(false, a, false, b2,
                                                   (short)0, acc[2], false, false);
  }

  const int rowbase = blockIdx.x * 16 + 8 * half;     // C/D: vgpr v -> row v+8*half
  const bool full = (blockIdx.x * 16 + 16) <= nrows;  // uniform branch
  #pragma unroll
  for (int t = 0; t < 3; ++t) {
    const int col = idx + 16 * t;
    if (col < FDIM) {
      const float bv = bias ? bias[col] : 0.0f;
      float* o = out + (size_t)rowbase * FDIM + col;
      if (full) {
        #pragma unroll
        for (int v = 0; v < 8; ++v) o[(size_t)v * FDIM] = acc[t][v] + bv;
      } else {
        #pragma unroll
        for (int v = 0; v < 8; ++v)
          if (rowbase + v < nrows) o[(size_t)v * FDIM] = acc[t][v] + bv;
      }
    }
  }
}

// el[n,h] = sum_d feat[n,h,d]*a_l[h,d]; er likewise. float4-vectorized.
__global__ void attn_scores(const float* __restrict__ feat,
                            const float* __restrict__ a_l,
                            const float* __restrict__ a_r,
                            float* __restrict__ el, float* __restrict__ er,
                            int n) {
  int gid = blockIdx.x * blockDim.x + threadIdx.x;
  if (gid >= n * NHEAD) return;
  const int node = gid / NHEAD;
  const int h    = gid % NHEAD;
  const float4* f4 = (const float4*)(feat + (size_t)node * FDIM + h * HDIM);
  const float4* l4 = (const float4*)(a_l + h * HDIM);
  const float4* r4 = (const float4*)(a_r + h * HDIM);
  float sl = 0.0f, sr = 0.0f;
  #pragma unroll
  for (int q = 0; q < HDIM / 4; ++q) {
    const float4 fv = f4[q], lv = l4[q], rv = r4[q];
    sl += fv.x * lv.x + fv.y * lv.y + fv.z * lv.z + fv.w * lv.w;
    sr += fv.x * rv.x + fv.y * rv.y + fv.z * rv.z + fv.w * rv.w;
  }
  el[gid] = sl;
  er[gid] = sr;
}

// rst=0 (n*36), mkey=-inf key, denom=0 (n*3).
__global__ void layer_init(unsigned* __restrict__ mkey,
                           float* __restrict__ denom,
                           float* __restrict__ rst, int n) {
  int gid = blockIdx.x * blockDim.x + threadIdx.x;
  if (gid < n * FDIM) rst[gid] = 0.0f;
  if (gid < n * NHEAD) {
    mkey[gid] = NEG_INF_KEY;
    denom[gid] = 0.0f;
  }
}

__global__ void edge_max(const int* __restrict__ src,
                         const int* __restrict__ dst,
                         const float* __restrict__ el,
                         const float* __restrict__ er,
                         unsigned* __restrict__ mkey, int ne) {
  int e = blockIdx.x * blockDim.x + threadIdx.x;
  if (e >= ne) return;
  const int s = src[e], d = dst[e];
  #pragma unroll
  for (int h = 0; h < NHEAD; ++h) {
    const float v = leaky(el[s * NHEAD + h] + er[d * NHEAD + h]);
    atomicMax(&mkey[d * NHEAD + h], f2key(v));
  }
}

// In-place: convert ordered keys back to floats; zero-degree guard (m finite).
__global__ void m_finalize(unsigned* __restrict__ mkey, float* __restrict__ m,
                           int n3) {
  int gid = blockIdx.x * blockDim.x + threadIdx.x;
  if (gid >= n3) return;
  float v = key2f(mkey[gid]);
  if (!isfinite(v)) v = 0.0f;
  m[gid] = v;
}

__global__ void edge_sum(const int* __restrict__ src,
                         const int* __restrict__ dst,
                         const float* __restrict__ el,
                         const float* __restrict__ er,
                         const float* __restrict__ m,
                         float* __restrict__ denom, int ne) {
  int e = blockIdx.x * blockDim.x + threadIdx.x;
  if (e >= ne) return;
  const int s = src[e], d = dst[e];
  #pragma unroll
  for (int h = 0; h < NHEAD; ++h) {
    const float v = leaky(el[s * NHEAD + h] + er[d * NHEAD + h]);
    const float ee = expf(v - m[d * NHEAD + h]);
    atomicAdd(&denom[d * NHEAD + h], ee);
  }
}

__global__ void edge_msg(const int* __restrict__ src,
                         const int* __restrict__ dst,
                         const float* __restrict__ el,
                         const float* __restrict__ er,
                         const float* __restrict__ m,
                         const float* __restrict__ denom,
                         const float* __restrict__ feat,
                         float* __restrict__ rst, int ne) {
  int e = blockIdx.x * blockDim.x + threadIdx.x;
  if (e >= ne) return;
  const int s = src[e], d = dst[e];
  const float4* fsrc4 = (const float4*)(feat + (size_t)s * FDIM); // 9 x b128
  __builtin_prefetch(fsrc4, 0, 3);   // global_prefetch_b8: warm gathered row
  #pragma unroll
  for (int h = 0; h < NHEAD; ++h) {
    const float v = leaky(el[s * NHEAD + h] + er[d * NHEAD + h]);
    const float alpha = expf(v - m[d * NHEAD + h]) /
                        fmaxf(denom[d * NHEAD + h], 1e-9f);
    float* rd = rst + (size_t)d * FDIM + h * HDIM;
    #pragma unroll
    for (int q = 0; q < HDIM / 4; ++q) {
      const float4 fv = fsrc4[h * (HDIM / 4) + q];
      atomicAdd(&rd[q * 4 + 0], fv.x * alpha);
      atomicAdd(&rd[q * 4 + 1], fv.y * alpha);
      atomicAdd(&rd[q * 4 + 2], fv.z * alpha);
      atomicAdd(&rd[q * 4 + 3], fv.w * alpha);
    }
  }
}

// h_next = ELU(rst + h + bias), in place over rst.
__global__ void node_finalize(const float* __restrict__ hin,
                              const float* __restrict__ bias,
                              float* __restrict__ rst, int n) {
  int gid = blockIdx.x * blockDim.x + threadIdx.x;
  if (gid >= n * FDIM) return;
  const float x = rst[gid] + hin[gid] + bias[gid % FDIM];
  rst[gid] = x > 0.0f ? x : expm1f(x);
}

extern "C" void kernel_launch(void* const* d_in, const int* in_sizes, int n_in,
                              void* d_out, int out_size, void* d_ws,
                              size_t ws_size, hipStream_t stream) {
  (void)n_in; (void)out_size; (void)ws_size;
  const float* X        = (const float*)d_in[0];
  const int*   src      = (const int*)d_in[1];
  const int*   dst      = (const int*)d_in[2];
  const float* lin0_w   = (const float*)d_in[3];
  const float* lin0_b   = (const float*)d_in[4];
  const float* fc_w     = (const float*)d_in[5];
  const float* attn_l   = (const float*)d_in[6];
  const float* attn_r   = (const float*)d_in[7];
  const float* gat_bias = (const float*)d_in[8];
  const float* out_w    = (const float*)d_in[9];
  const float* out_b    = (const float*)d_in[10];

  const int n  = in_sizes[0] / FDIM;
  const int ne = in_sizes[1];

  // Workspace carve-up: N*120 floats (~48 MB for N=100000).
  float* ws   = (float*)d_ws;
  float* hA    = ws; ws += (size_t)n * FDIM;
  float* hB    = ws; ws += (size_t)n * FDIM;
  float* feat  = ws; ws += (size_t)n * FDIM;
  float* el    = ws; ws += (size_t)n * NHEAD;
  float* er    = ws; ws += (size_t)n * NHEAD;
  float* mbuf  = ws; ws += (size_t)n * NHEAD;  // uint keys, then floats in place
  float* denom = ws; ws += (size_t)n * NHEAD;

  const dim3 gemmGrid((n + 15) / 16);
  const int nbE  = (ne + 255) / 256;
  const int nbN3 = (n * NHEAD + 255) / 256;
  const int nbNF = (n * FDIM + 255) / 256;

  // h = X @ lin0_w + lin0_b
  gemm36_wmma<<<gemmGrid, 32, 0, stream>>>(X, lin0_w, lin0_b, hA, n);

  float* hcur = hA;
  float* hnxt = hB;
  for (int l = 0; l < 3; ++l) {
    gemm36_wmma<<<gemmGrid, 32, 0, stream>>>(
        hcur, fc_w + (size_t)l * FDIM * FDIM, nullptr, feat, n);
    attn_scores<<<nbN3, 256, 0, stream>>>(
        feat, attn_l + (size_t)l * NHEAD * HDIM, attn_r + (size_t)l * NHEAD * HDIM,
        el, er, n);
    layer_init<<<nbNF, 256, 0, stream>>>((unsigned*)mbuf, denom, hnxt, n);
    edge_max<<<nbE, 256, 0, stream>>>(src, dst, el, er, (unsigned*)mbuf, ne);
    m_finalize<<<nbN3, 256, 0, stream>>>((unsigned*)mbuf, mbuf, n * NHEAD);
    edge_sum<<<nbE, 256, 0, stream>>>(src, dst, el, er, mbuf, denom, ne);
    edge_msg<<<nbE, 256, 0, stream>>>(src, dst, el, er, mbuf, denom, feat,
                                      hnxt, ne);
    node_finalize<<<nbNF, 256, 0, stream>>>(hcur, gat_bias + (size_t)l * FDIM,
                                            hnxt, n);
    float* t = hcur; hcur = hnxt; hnxt = t;
  }

  // out = h @ out_w + out_b
  gemm36_wmma<<<gemmGrid, 32, 0, stream>>>(hcur, out_w, out_b, (float*)d_out, n);
}